// HashEncoding_6038724018404
// MI455X (gfx1250) — compile-verified
//
#include <hip/hip_runtime.h>

// Instant-NGP style multilevel hash encoding for MI455X (gfx1250).
// 16 levels, TABLE_MOD = 4096 entries x 2 feats per level (32 KB/level).
// Strategy: split levels into 2 groups of 8; each workgroup stages its
// group's 256 KB of tables into LDS (CDNA5: up to 320 KB LDS per workgroup)
// using async global->LDS copies when available, then grid-strides over
// points doing LDS gathers (ds_load_b64) and nontemporal 128-bit stores of
// its contiguous 64-byte half-row.

#ifndef __has_builtin
#define __has_builtin(x) 0
#endif

#if __has_builtin(__builtin_amdgcn_global_load_async_to_lds_b128)
#define USE_ASYNC_LDS_PRELOAD 1
#else
#define USE_ASYNC_LDS_PRELOAD 0
#endif

namespace {

typedef __attribute__((ext_vector_type(4))) float v4f;
typedef __attribute__((ext_vector_type(2))) float v2f;
typedef __attribute__((ext_vector_type(4))) int   v4i;

typedef __attribute__((address_space(1))) v4i* gv4i_ptr;  // global
typedef __attribute__((address_space(3))) v4i* lv4i_ptr;  // LDS

constexpr int kRes[16] = {16, 22, 30, 42, 58, 80, 111, 153,
                          212, 294, 406, 561, 776, 1072, 1482, 2048};
constexpr unsigned kP1 = 2481u;  // 2654435761 % 4096
constexpr unsigned kP2 = 1941u;  // 805459861  % 4096
constexpr int kTableMod = 4096;
constexpr int kLevelsPerGroup = 8;
constexpr int kThreads = 1024;   // 32 wave32s per workgroup
constexpr int kBlocks = 512;     // persistent-ish; one 256KB-LDS block per WGP

__device__ inline void wait_asynccnt0() {
#if __has_builtin(__builtin_amdgcn_s_wait_asynccnt)
    __builtin_amdgcn_s_wait_asynccnt(0);
#else
    asm volatile("s_wait_asynccnt 0x0" ::: "memory");
#endif
}

template <int GROUP>
__global__ __launch_bounds__(kThreads, 1)
void hashenc_kernel(const float* __restrict__ pos,
                    const float* __restrict__ tables,
                    float* __restrict__ out,
                    int npoints)
{
    // 8 levels * 4096 entries * float2 = 256 KB (legal only on CDNA5)
    __shared__ v2f lut[kLevelsPerGroup * kTableMod];

    // Cooperative preload of this group's tables into LDS.
    {
        const v4f* __restrict__ src =
            reinterpret_cast<const v4f*>(tables) +
            (size_t)GROUP * kLevelsPerGroup * (kTableMod / 2);
        v4f* dst = reinterpret_cast<v4f*>(lut);
        constexpr int n4 = kLevelsPerGroup * kTableMod / 2;  // 16384 v4f
#if USE_ASYNC_LDS_PRELOAD
        // Direct global->LDS copies, tracked with ASYNCcnt (no VGPR round-trip).
        for (int i = (int)threadIdx.x; i < n4; i += kThreads) {
            __builtin_amdgcn_global_load_async_to_lds_b128(
                (gv4i_ptr)(src + i),
                (lv4i_ptr)(dst + i),
                /*offset=*/0, /*cpol=*/0);
        }
        wait_asynccnt0();
#else
        for (int i = (int)threadIdx.x; i < n4; i += kThreads)
            dst[i] = src[i];
#endif
    }
    __syncthreads();

    const int stride = (int)gridDim.x * kThreads;
    for (int n = (int)blockIdx.x * kThreads + (int)threadIdx.x;
         n < npoints; n += stride) {
        // Streaming reads: keep out of caches (tables stay hot in L2/LDS).
        const float px = __builtin_nontemporal_load(pos + 3 * n + 0);
        const float py = __builtin_nontemporal_load(pos + 3 * n + 1);
        const float pz = __builtin_nontemporal_load(pos + 3 * n + 2);

        v2f f[kLevelsPerGroup];
#pragma unroll
        for (int j = 0; j < kLevelsPerGroup; ++j) {
            const int res = kRes[GROUP * kLevelsPerGroup + j];
            const float m = (float)(res - 1);
            // positions are in [0,1): truncation == floor (nonnegative)
            int cx = (int)(px * m);
            int cy = (int)(py * m);
            int cz = (int)(pz * m);
            cx = min(max(cx, 0), res - 1);
            cy = min(max(cy, 0), res - 1);
            cz = min(max(cz, 0), res - 1);
            const unsigned h =
                ((unsigned)cx + (unsigned)cy * kP1 + (unsigned)cz * kP2) & 4095u;
            f[j] = lut[j * kTableMod + (int)h];  // ds_load_b64 gather
        }

        // Contiguous 64 B half-row, 64 B aligned: 4x nontemporal b128 stores.
        float* o = out + (size_t)n * 32 + GROUP * 16;
        v4f* o4 = reinterpret_cast<v4f*>(o);
        v4f v0 = {f[0].x, f[0].y, f[1].x, f[1].y};
        v4f v1 = {f[2].x, f[2].y, f[3].x, f[3].y};
        v4f v2 = {f[4].x, f[4].y, f[5].x, f[5].y};
        v4f v3 = {f[6].x, f[6].y, f[7].x, f[7].y};
        __builtin_nontemporal_store(v0, o4 + 0);
        __builtin_nontemporal_store(v1, o4 + 1);
        __builtin_nontemporal_store(v2, o4 + 2);
        __builtin_nontemporal_store(v3, o4 + 3);
    }
}

}  // namespace

extern "C" void kernel_launch(void* const* d_in, const int* in_sizes, int n_in,
                              void* d_out, int out_size, void* d_ws, size_t ws_size,
                              hipStream_t stream) {
    (void)n_in; (void)out_size; (void)d_ws; (void)ws_size;
    const float* pos    = (const float*)d_in[0];  // [N,3] f32
    const float* tables = (const float*)d_in[1];  // [16,4096,2] f32
    float* out          = (float*)d_out;          // [N,32] f32
    const int npoints = in_sizes[0] / 3;

    hashenc_kernel<0><<<dim3(kBlocks), dim3(kThreads), 0, stream>>>(pos, tables, out, npoints);
    hashenc_kernel<1><<<dim3(kBlocks), dim3(kThreads), 0, stream>>>(pos, tables, out, npoints);
}